// GCMCLayer_22497038697226
// MI455X (gfx1250) — compile-verified
//
#include <hip/hip_runtime.h>

typedef float v2f __attribute__((ext_vector_type(2)));
typedef float v8f __attribute__((ext_vector_type(8)));

#define N_USER  100000
#define N_ITEM  50000
#define R_ETYPE 5
#define E_EDGE  500000
#define D_IN    512
#define M_UNITS 128
#define MSG     640     // R_ETYPE * M_UNITS
#define OUT_N   256

// ---------------------------------------------------------------------------
// Batched transpose: in [B][K][N] -> out [B][N][K]. Tiny (<= 1.25 MB), runs once
// per call so GEMM B-fragments become contiguous b64 loads.
// ---------------------------------------------------------------------------
__global__ void transpose_kernel(const float* __restrict__ in,
                                 float* __restrict__ out,
                                 int K, int N, int total)
{
    const int idx = blockIdx.x * blockDim.x + threadIdx.x;
    if (idx >= total) return;
    const int n = idx % N;
    const int k = (idx / N) % K;
    const int b = idx / (N * K);
    out[((size_t)b * N + n) * K + k] = in[idx];
}

// ---------------------------------------------------------------------------
// h[n][r*128+j] = (feat[n] . W[r][:,j]) * cj[n]
// Wave computes a 64x16 tile: 4 M-subtiles share one B fragment per k-step.
// B comes pre-transposed: Wt[r][j][k]  (contiguous in k).
// ---------------------------------------------------------------------------
__global__ void gemm_h_kernel(const float* __restrict__ feat,  // [N][512]
                              const float* __restrict__ Wt,    // [R][128][512]
                              const float* __restrict__ cj,    // [N]
                              float* __restrict__ h,           // [N][640]
                              int nrows)
{
    const int lane = threadIdx.x;        // 0..31
    const int half = lane >> 4;          // 0/1
    const int l16  = lane & 15;
    const int rowBase64 = blockIdx.x * 64;
    const int colTile = blockIdx.y * blockDim.y + threadIdx.y;  // 0..39
    const int jGlobal = colTile * 16;
    const int r  = jGlobal / M_UNITS;
    const int jW = jGlobal % M_UNITS;

    // B fragment source: Wt[r][jW + l16][*]
    const float* __restrict__ bRow =
        Wt + ((size_t)r * M_UNITS + jW + l16) * D_IN;

    int rb[4];
    const float* aRow[4];
#pragma unroll
    for (int t = 0; t < 4; ++t) {
        rb[t] = rowBase64 + 16 * t;
        const int rc = rb[t] > nrows - 16 ? nrows - 16 : rb[t];  // clamp tail
        aRow[t] = feat + (size_t)(rc + l16) * D_IN;
    }

    v8f acc0 = {}, acc1 = {}, acc2 = {}, acc3 = {};
#pragma unroll 4
    for (int k = 0; k < D_IN; k += 4) {
        const int ka = k + 2 * half;
        const v2f b  = *(const v2f*)(bRow + ka);     // B[ka][n], B[ka+1][n]
        const v2f a0 = *(const v2f*)(aRow[0] + ka);
        const v2f a1 = *(const v2f*)(aRow[1] + ka);
        const v2f a2 = *(const v2f*)(aRow[2] + ka);
        const v2f a3 = *(const v2f*)(aRow[3] + ka);
        acc0 = __builtin_amdgcn_wmma_f32_16x16x4_f32(false, a0, false, b, (short)0, acc0, false, false);
        acc1 = __builtin_amdgcn_wmma_f32_16x16x4_f32(false, a1, false, b, (short)0, acc1, false, false);
        acc2 = __builtin_amdgcn_wmma_f32_16x16x4_f32(false, a2, false, b, (short)0, acc2, false, false);
        acc3 = __builtin_amdgcn_wmma_f32_16x16x4_f32(false, a3, false, b, (short)0, acc3, false, false);
    }

    v8f accs[4] = {acc0, acc1, acc2, acc3};
#pragma unroll
    for (int t = 0; t < 4; ++t) {
        if (rb[t] <= nrows - 16) {
#pragma unroll
            for (int g = 0; g < 8; ++g) {
                const int row = rb[t] + g + 8 * half;      // C/D VGPR layout
                h[(size_t)row * MSG + jGlobal + l16] = accs[t][g] * cj[row];
            }
        }
    }
}

// ---------------------------------------------------------------------------
// out[n][c] = ci[n] * (agg[n] . fcw[:,c]) + fcb[c]   (fcw pre-transposed)
// ---------------------------------------------------------------------------
__global__ void gemm_fc_kernel(const float* __restrict__ agg,   // [N][640]
                               const float* __restrict__ ci,    // [N]
                               const float* __restrict__ fcwt,  // [256][640]
                               const float* __restrict__ fcb,   // [256]
                               float* __restrict__ out,         // [N][256]
                               int nrows)
{
    const int lane = threadIdx.x;
    const int half = lane >> 4;
    const int l16  = lane & 15;
    const int rowBase64 = blockIdx.x * 64;
    const int colTile = blockIdx.y * blockDim.y + threadIdx.y;  // 0..15
    const int n0 = colTile * 16;

    const float* __restrict__ bRow = fcwt + (size_t)(n0 + l16) * MSG;

    int rb[4];
    const float* aRow[4];
#pragma unroll
    for (int t = 0; t < 4; ++t) {
        rb[t] = rowBase64 + 16 * t;
        const int rc = rb[t] > nrows - 16 ? nrows - 16 : rb[t];
        aRow[t] = agg + (size_t)(rc + l16) * MSG;
    }

    v8f acc0 = {}, acc1 = {}, acc2 = {}, acc3 = {};
#pragma unroll 4
    for (int k = 0; k < MSG; k += 4) {
        const int ka = k + 2 * half;
        const v2f b  = *(const v2f*)(bRow + ka);
        const v2f a0 = *(const v2f*)(aRow[0] + ka);
        const v2f a1 = *(const v2f*)(aRow[1] + ka);
        const v2f a2 = *(const v2f*)(aRow[2] + ka);
        const v2f a3 = *(const v2f*)(aRow[3] + ka);
        acc0 = __builtin_amdgcn_wmma_f32_16x16x4_f32(false, a0, false, b, (short)0, acc0, false, false);
        acc1 = __builtin_amdgcn_wmma_f32_16x16x4_f32(false, a1, false, b, (short)0, acc1, false, false);
        acc2 = __builtin_amdgcn_wmma_f32_16x16x4_f32(false, a2, false, b, (short)0, acc2, false, false);
        acc3 = __builtin_amdgcn_wmma_f32_16x16x4_f32(false, a3, false, b, (short)0, acc3, false, false);
    }

    const float bias = fcb[n0 + l16];
    v8f accs[4] = {acc0, acc1, acc2, acc3};
#pragma unroll
    for (int t = 0; t < 4; ++t) {
        if (rb[t] <= nrows - 16) {
#pragma unroll
            for (int g = 0; g < 8; ++g) {
                const int row = rb[t] + g + 8 * half;
                out[(size_t)row * OUT_N + n0 + l16] = accs[t][g] * ci[row] + bias;
            }
        }
    }
}

// ---------------------------------------------------------------------------
// Edge phase: per (r, e): agg_i[dst][r*128+:128] += hu[src][r*128+:128]
//                         agg_u[src][r*128+:128] += hi[dst][r*128+:128]
// 64 threads per edge: 32 lanes/direction, float4 read + 4 float atomics.
// ---------------------------------------------------------------------------
__global__ void edge_scatter_kernel(const int* __restrict__ src_u,   // [R][E]
                                    const int* __restrict__ dst_i,   // [R][E]
                                    const float* __restrict__ hu,    // [Nu][640]
                                    const float* __restrict__ hi,    // [Ni][640]
                                    float* __restrict__ agg_u,       // [Nu][640]
                                    float* __restrict__ agg_i)       // [Ni][640]
{
    const int r = blockIdx.y;
    const int e = blockIdx.x * 4 + (threadIdx.x >> 6);
    if (e >= E_EDGE) return;
    const int dir  = (threadIdx.x >> 5) & 1;
    const int lane = threadIdx.x & 31;

    const int s = src_u[(size_t)r * E_EDGE + e];
    const int d = dst_i[(size_t)r * E_EDGE + e];

    const float4* __restrict__ src;
    float* __restrict__ dst;
    if (dir == 0) {   // user -> item
        src = (const float4*)(hu + (size_t)s * MSG + r * M_UNITS);
        dst = agg_i + (size_t)d * MSG + r * M_UNITS;
    } else {          // item -> user (reversed edges)
        src = (const float4*)(hi + (size_t)d * MSG + r * M_UNITS);
        dst = agg_u + (size_t)s * MSG + r * M_UNITS;
    }

    const float4 v = src[lane];
    const int c = lane * 4;
    atomicAdd(dst + c + 0, v.x);
    atomicAdd(dst + c + 1, v.y);
    atomicAdd(dst + c + 2, v.z);
    atomicAdd(dst + c + 3, v.w);
}

// ---------------------------------------------------------------------------
extern "C" void kernel_launch(void* const* d_in, const int* in_sizes, int n_in,
                              void* d_out, int out_size, void* d_ws, size_t ws_size,
                              hipStream_t stream)
{
    (void)in_sizes; (void)n_in; (void)out_size; (void)ws_size;

    const float* ufeat   = (const float*)d_in[0];
    const float* ifeat   = (const float*)d_in[1];
    const float* W_ui    = (const float*)d_in[2];
    const float* W_iu    = (const float*)d_in[3];
    const float* ufc_w   = (const float*)d_in[4];
    const float* ufc_b   = (const float*)d_in[5];
    const float* ifc_w   = (const float*)d_in[6];
    const float* ifc_b   = (const float*)d_in[7];
    const float* cj_user = (const float*)d_in[8];
    const float* ci_user = (const float*)d_in[9];
    const float* cj_item = (const float*)d_in[10];
    const float* ci_item = (const float*)d_in[11];
    const int*   src_u   = (const int*)d_in[12];
    const int*   dst_i   = (const int*)d_in[13];

    // Workspace layout (floats):
    //   hu | hi | agg_u | agg_i | Wt_ui | Wt_iu | ufc_wt | ifc_wt
    float* hu     = (float*)d_ws;
    float* hi     = hu     + (size_t)N_USER * MSG;
    float* agg_u  = hi     + (size_t)N_ITEM * MSG;
    float* agg_i  = agg_u  + (size_t)N_USER * MSG;
    float* Wt_ui  = agg_i  + (size_t)N_ITEM * MSG;
    float* Wt_iu  = Wt_ui  + (size_t)R_ETYPE * D_IN * M_UNITS;
    float* ufc_wt = Wt_iu  + (size_t)R_ETYPE * D_IN * M_UNITS;
    float* ifc_wt = ufc_wt + (size_t)MSG * OUT_N;

    // Zero the scatter accumulators every call (capture-safe memset node).
    hipMemsetAsync(agg_u, 0,
                   (size_t)(N_USER + N_ITEM) * MSG * sizeof(float), stream);

    // Stage transposed weights (tiny: 1.25MB + 1.25MB + 640KB + 640KB).
    {
        const int totW  = R_ETYPE * D_IN * M_UNITS;   // 327680
        const int totFC = MSG * OUT_N;                // 163840
        transpose_kernel<<<(totW  + 255) / 256, 256, 0, stream>>>(W_ui,  Wt_ui,  D_IN, M_UNITS, totW);
        transpose_kernel<<<(totW  + 255) / 256, 256, 0, stream>>>(W_iu,  Wt_iu,  D_IN, M_UNITS, totW);
        transpose_kernel<<<(totFC + 255) / 256, 256, 0, stream>>>(ufc_w, ufc_wt, MSG,  OUT_N,   totFC);
        transpose_kernel<<<(totFC + 255) / 256, 256, 0, stream>>>(ifc_w, ifc_wt, MSG,  OUT_N,   totFC);
    }

    const dim3 blk(32, 4, 1);   // 4 waves / block, one 64x16 tile per wave

    // Phase 1: per-rating feature GEMMs (stacked 640-col layout)
    gemm_h_kernel<<<dim3((N_USER + 63) / 64, (MSG / 16) / 4), blk, 0, stream>>>(
        ufeat, Wt_ui, cj_user, hu, N_USER);
    gemm_h_kernel<<<dim3((N_ITEM + 63) / 64, (MSG / 16) / 4), blk, 0, stream>>>(
        ifeat, Wt_iu, cj_item, hi, N_ITEM);

    // Phase 2: edge gather + scatter-add (both directions)
    edge_scatter_kernel<<<dim3((E_EDGE + 3) / 4, R_ETYPE), dim3(256), 0, stream>>>(
        src_u, dst_i, hu, hi, agg_u, agg_i);

    // Phase 3: output FCs (ci + bias fused in epilogue)
    float* u_out = (float*)d_out;
    float* i_out = u_out + (size_t)N_USER * OUT_N;
    gemm_fc_kernel<<<dim3((N_USER + 63) / 64, (OUT_N / 16) / 4), blk, 0, stream>>>(
        agg_u, ci_user, ufc_wt, ufc_b, u_out, N_USER);
    gemm_fc_kernel<<<dim3((N_ITEM + 63) / 64, (OUT_N / 16) / 4), blk, 0, stream>>>(
        agg_i, ci_item, ifc_wt, ifc_b, i_out, N_ITEM);
}